// K_MOTE_84026740179071
// MI455X (gfx1250) — compile-verified
//
#include <hip/hip_runtime.h>
#include <hip/hip_bf16.h>

typedef float v2f __attribute__((ext_vector_type(2)));
typedef float v4f __attribute__((ext_vector_type(4)));
typedef float v8f __attribute__((ext_vector_type(8)));

#define GAMMA 4.0f
#define WSCALE_INV 2.0f   /* 1 / 0.5 */

__device__ __forceinline__ v8f wmma4(v2f a, v2f b, v8f c) {
    // D = A(16x4 f32) * B(4x16 f32) + C(16x16 f32), wave32, full f32 precision
    return __builtin_amdgcn_wmma_f32_16x16x4_f32(
        /*neg_a=*/false, a, /*neg_b=*/false, b,
        /*c_mod=*/(short)0, c, /*reuse_a=*/false, /*reuse_b=*/false);
}

__global__ __launch_bounds__(256) void kan_moe_kernel(
    const float* __restrict__ x_in,
    const float* __restrict__ rw1, const float* __restrict__ rb1,
    const float* __restrict__ rw2, const float* __restrict__ rb2,
    const float* __restrict__ Wf, const float* __restrict__ Ws,
    const float* __restrict__ Wr, const float* __restrict__ Ww,
    float* __restrict__ out_emb, float* __restrict__ out_raw,
    float* __restrict__ out_mask, int Btot)
{
    __shared__ float s_w1[32], s_b1[32], s_w2[128], s_b2[4];
    const int tid = threadIdx.x;
    if (tid < 32)  { s_w1[tid] = rw1[tid]; s_b1[tid] = rb1[tid]; }
    if (tid < 128) { s_w2[tid] = rw2[tid]; }
    if (tid < 4)   { s_b2[tid] = rb2[tid]; }
    __syncthreads();

    const int lane = tid & 31;
    const int wave = tid >> 5;
    const int hi   = lane >> 4;   // which half of the wave (K split)
    const int lo   = lane & 15;   // M index (A) / N index (B,C)

    // ---- Preload expert weights as WMMA B-fragments (lane l, vgpr v holds
    //      W[k][n] with k = 4*chunk + 2*hi + v, n = lo + 16*ntile) ----
    v2f wf[8][2], wsp[4][2], wrk[4][2], wwv[4][2];
    #pragma unroll
    for (int c = 0; c < 8; ++c)
        #pragma unroll
        for (int t = 0; t < 2; ++t) {
            const int k = 4*c + 2*hi, n = lo + 16*t;
            v2f w = { Wf[(k+0)*32 + n], Wf[(k+1)*32 + n] };
            wf[c][t] = w;
        }
    #pragma unroll
    for (int c = 0; c < 4; ++c)
        #pragma unroll
        for (int t = 0; t < 2; ++t) {
            const int k = 4*c + 2*hi, n = lo + 16*t;
            v2f a = { Ws[(k+0)*32 + n], Ws[(k+1)*32 + n] };
            v2f b = { Wr[(k+0)*32 + n], Wr[(k+1)*32 + n] };
            v2f d = { Ww[(k+0)*32 + n], Ww[(k+1)*32 + n] };
            wsp[c][t] = a; wrk[c][t] = b; wwv[c][t] = d;
        }

    const long nTiles = (long)Btot >> 5;              // 32 rows per wave-tile
    const long waveId = (long)blockIdx.x * 8 + wave;
    const long nWaves = (long)gridDim.x * 8;

    for (long tile = waveId; tile < nTiles; tile += nWaves) {
        const long base = tile << 5;
        const long row  = base + lane;

        // prefetch the next tile's timestamps (speculative is safe)
        __builtin_prefetch(x_in + ((tile + nWaves) << 5), 0, 1);

        // ---------------- Router MLP + softmax + top-2 (one row per lane) ---
        const float x = x_in[row];
        float lg[4] = { s_b2[0], s_b2[1], s_b2[2], s_b2[3] };
        #pragma unroll
        for (int j = 0; j < 32; ++j) {
            const float h = fmaxf(fmaf(x, s_w1[j], s_b1[j]), 0.0f);
            lg[0] = fmaf(h, s_w2[4*j+0], lg[0]);
            lg[1] = fmaf(h, s_w2[4*j+1], lg[1]);
            lg[2] = fmaf(h, s_w2[4*j+2], lg[2]);
            lg[3] = fmaf(h, s_w2[4*j+3], lg[3]);
        }
        const float mx = fmaxf(fmaxf(lg[0], lg[1]), fmaxf(lg[2], lg[3]));
        float r[4], sum = 0.0f;
        #pragma unroll
        for (int e = 0; e < 4; ++e) { r[e] = __expf(lg[e] - mx); sum += r[e]; }
        const float inv = 1.0f / sum;
        #pragma unroll
        for (int e = 0; e < 4; ++e) r[e] *= inv;

        // top-2, lowest index wins ties (matches lax.top_k)
        int i0 = 0;
        #pragma unroll
        for (int e = 1; e < 4; ++e) if (r[e] > r[i0]) i0 = e;
        int i1 = (i0 == 0) ? 1 : 0;
        #pragma unroll
        for (int e = 0; e < 4; ++e) if (e != i0 && r[e] > r[i1]) i1 = e;

        float dw[4], mk[4];
        #pragma unroll
        for (int e = 0; e < 4; ++e) {
            const bool sel = (e == i0) | (e == i1);
            dw[e] = sel ? r[e] : 0.0f;
            mk[e] = sel ? 1.0f : 0.0f;
        }

        v4f r4 = { r[0], r[1], r[2], r[3] };
        v4f m4 = { mk[0], mk[1], mk[2], mk[3] };
        *(v4f*)(out_raw  + row * 4) = r4;
        *(v4f*)(out_mask + row * 4) = m4;

        // ---------------- Expert matmuls: two 16-row WMMA sub-tiles ---------
        #pragma unroll
        for (int s = 0; s < 2; ++s) {
            const float xm = x_in[base + s*16 + lo];  // A-fragment row for this lane

            v8f af0 = {}, af1 = {}, as0 = {}, as1 = {};
            v8f ar0 = {}, ar1 = {}, aw0 = {}, aw1 = {};

            // Fourier expert: K = 32 (sin freqs 1..16, then cos freqs 1..16)
            #pragma unroll
            for (int c = 0; c < 8; ++c) {
                const int k0 = 4*c + 2*hi;
                v2f a;
                if (c < 4) {
                    a[0] = __sinf(xm * (float)(k0 + 1));
                    a[1] = __sinf(xm * (float)(k0 + 2));
                } else {
                    a[0] = __cosf(xm * (float)(k0 - 15));
                    a[1] = __cosf(xm * (float)(k0 - 14));
                }
                af0 = wmma4(a, wf[c][0], af0);
                af1 = wmma4(a, wf[c][1], af1);
            }

            // Spline / RKHS / Wavelet experts: K = 16, shared knots
            #pragma unroll
            for (int c = 0; c < 4; ++c) {
                const int k0 = 4*c + 2*hi;
                v2f a_s, a_r, a_w;
                #pragma unroll
                for (int v = 0; v < 2; ++v) {
                    const float kn = -2.0f + (4.0f / 15.0f) * (float)(k0 + v);
                    const float d  = xm - kn;
                    const float rl = fmaxf(d, 0.0f);
                    a_s[v] = rl * rl * rl;                       // relu(x-knot)^3
                    const float d2 = d * d;
                    a_r[v] = __expf(-GAMMA * d2);                // gaussian RBF
                    const float z2 = (WSCALE_INV * WSCALE_INV) * d2;
                    a_w[v] = (1.0f - z2) * __expf(-0.5f * z2);   // ricker
                }
                as0 = wmma4(a_s, wsp[c][0], as0);  as1 = wmma4(a_s, wsp[c][1], as1);
                ar0 = wmma4(a_r, wrk[c][0], ar0);  ar1 = wmma4(a_r, wrk[c][1], ar1);
                aw0 = wmma4(a_w, wwv[c][0], aw0);  aw1 = wmma4(a_w, wwv[c][1], aw1);
            }

            // ---- Scale by dispatch weights (cross-lane via ds_bpermute) and
            //      store: C layout row m = v + 8*hi, col n = lo (+16 per tile)
            #pragma unroll
            for (int v = 0; v < 8; ++v) {
                const int ridx = s*16 + v + 8*hi;           // row within 32-row tile
                const float d0 = __shfl(dw[0], ridx, 32);
                const float d1 = __shfl(dw[1], ridx, 32);
                const float d2 = __shfl(dw[2], ridx, 32);
                const float d3 = __shfl(dw[3], ridx, 32);
                float* o = out_emb + (base + ridx) * 128 + lo;
                o[  0] = af0[v] * d0;   o[ 16] = af1[v] * d0;   // fourier -> cols 0..31
                o[ 32] = as0[v] * d1;   o[ 48] = as1[v] * d1;   // spline  -> cols 32..63
                o[ 64] = ar0[v] * d2;   o[ 80] = ar1[v] * d2;   // rkhs    -> cols 64..95
                o[ 96] = aw0[v] * d3;   o[112] = aw1[v] * d3;   // wavelet -> cols 96..127
            }
        }
    }
}

extern "C" void kernel_launch(void* const* d_in, const int* in_sizes, int n_in,
                              void* d_out, int out_size, void* d_ws, size_t ws_size,
                              hipStream_t stream) {
    const float* x   = (const float*)d_in[0];
    const float* rw1 = (const float*)d_in[1];
    const float* rb1 = (const float*)d_in[2];
    const float* rw2 = (const float*)d_in[3];
    const float* rb2 = (const float*)d_in[4];
    const float* Wf  = (const float*)d_in[5];
    const float* Ws  = (const float*)d_in[6];
    const float* Wr  = (const float*)d_in[7];
    const float* Ww  = (const float*)d_in[8];

    const int Btot = in_sizes[0];              // 524288
    float* out      = (float*)d_out;
    float* out_emb  = out;                     // (B, 128)
    float* out_raw  = out + (long)Btot * 128;  // (B, 4)
    float* out_mask = out + (long)Btot * 132;  // (B, 4)

    const int blocks = 1024;                   // 8192 waves, grid-stride over 16384 tiles
    kan_moe_kernel<<<blocks, 256, 0, stream>>>(
        x, rw1, rb1, rw2, rb2, Wf, Ws, Wr, Ww,
        out_emb, out_raw, out_mask, Btot);
}